// QuantumHead_80925773791384
// MI455X (gfx1250) — compile-verified
//
#include <hip/hip_runtime.h>
#include <math.h>
#include <stdint.h>

typedef float v2f __attribute__((ext_vector_type(2)));
typedef float v8f __attribute__((ext_vector_type(8)));

#define PI_F 3.14159265358979323846f

// ---------------- 4-qubit gate helpers (state in registers, fully unrolled) --------
template<int W>
__device__ __forceinline__ void ry_gate(float* sr, float* si, float c, float s) {
  constexpr int M = 8 >> W;            // wire 0 = most significant index bit
  #pragma unroll
  for (int i = 0; i < 16; ++i) {
    if ((i & M) == 0) {
      const int j = i | M;
      float a0r = sr[i], a0i = si[i], a1r = sr[j], a1i = si[j];
      sr[i] = c * a0r - s * a1r;  si[i] = c * a0i - s * a1i;
      sr[j] = s * a0r + c * a1r;  si[j] = s * a0i + c * a1i;
    }
  }
}

template<int W>
__device__ __forceinline__ void rz_gate(float* sr, float* si, float pc, float ps) {
  // bit==0: multiply by (pc, -ps)=e^{-i t/2}; bit==1: (pc, +ps)
  constexpr int M = 8 >> W;
  #pragma unroll
  for (int i = 0; i < 16; ++i) {
    const float q = (i & M) ? ps : -ps;
    const float r = sr[i], m = si[i];
    sr[i] = r * pc - m * q;
    si[i] = r * q + m * pc;
  }
}

template<int C>
__device__ __forceinline__ void cnot_gate(float* sr, float* si) {
  constexpr int MC = 8 >> C;
  constexpr int MT = 8 >> (C + 1);
  #pragma unroll
  for (int i = 0; i < 16; ++i) {
    if ((i & MC) != 0 && (i & MT) == 0) {
      const int j = i | MT;
      float t;
      t = sr[i]; sr[i] = sr[j]; sr[j] = t;
      t = si[i]; si[i] = si[j]; si[j] = t;
    }
  }
}

// async global->LDS 16B DMA (CDNA5; ASYNCcnt-tracked, no VGPR round trip)
__device__ __forceinline__ void async_copy_b128(unsigned lds_byte_addr,
                                                const float* gptr) {
  asm volatile("global_load_async_to_lds_b128 %0, %1, off"
               :: "v"(lds_byte_addr), "v"(gptr)
               : "memory");
}
__device__ __forceinline__ void wait_async0() {
  asm volatile("s_wait_asynccnt 0x0" ::: "memory");
}

// ---------------- fused kernel ----------------------------------------------------
__global__ __launch_bounds__(256)
void QuantumHead_80925773791384_kernel(
    const float* __restrict__ x,      // [B,256]
    const float* __restrict__ w_proj, // [4,256]
    const float* __restrict__ b_proj, // [4]
    const float* __restrict__ qw,     // [2,4,2]
    const float* __restrict__ w_h,    // [16,4]
    const float* __restrict__ b_h,    // [16]
    const float* __restrict__ w_out,  // [10,16]
    const float* __restrict__ b_out,  // [10]
    float* __restrict__ out,          // [B,10]
    int Btot) {
  __shared__ float x_lds[256 * 36];    // x K-chunk; reused for output staging
  __shared__ float wp_lds[16 * 260];   // w_proj zero-padded to 16 rows
  __shared__ float proj_lds[256 * 4];
  __shared__ float qtrig[32];          // {cos,sin}(qw/2) for 16 gate params

  const int tid  = threadIdx.x;
  const int lane = tid & 31;
  const int wave = tid >> 5;
  const int hf   = lane >> 4;          // 0 or 1 (half-wave)
  const int mn   = lane & 15;          // M (A-frag row) / N (B,D column)
  const int row0 = blockIdx.x * 256;
  const bool full = (row0 + 256 <= Btot);   // block-uniform

  for (int idx = tid; idx < 16 * 260; idx += 256) {
    const int r = idx / 260, c = idx - r * 260;
    wp_lds[idx] = (r < 4 && c < 256) ? w_proj[r * 256 + c] : 0.0f;
  }
  if (tid < 16) {
    float s, c;
    sincosf(0.5f * qw[tid], &s, &c);
    qtrig[tid * 2 + 0] = c;
    qtrig[tid * 2 + 1] = s;
  }
  __syncthreads();

  // ---- projection: 16x16 WMMA tiles, 2 tiles per wave, K=256 in chunks of 32 ----
  v8f acc0 = {0.f, 0.f, 0.f, 0.f, 0.f, 0.f, 0.f, 0.f};
  v8f acc1 = {0.f, 0.f, 0.f, 0.f, 0.f, 0.f, 0.f, 0.f};

  const int lr   = tid >> 3;           // 0..31: row-chunk owner
  const int part = tid & 7;            // 0..7 : float4 within the 128B row-chunk

  #pragma unroll 1
  for (int kc = 0; kc < 8; ++kc) {     // K chunks of 32
    if (!full) {                       // rare tail block: pre-zero the tile
      for (int idx = tid; idx < 256 * 36; idx += 256) x_lds[idx] = 0.f;
      __syncthreads();
    }
    #pragma unroll
    for (int i = 0; i < 8; ++i) {
      const int lrow = lr + i * 32;
      if (full || (row0 + lrow < Btot)) {
        const unsigned laddr =
            (unsigned)(uintptr_t)(&x_lds[lrow * 36 + part * 4]);
        const float* gp =
            x + (size_t)(row0 + lrow) * 256 + kc * 32 + part * 4;
        async_copy_b128(laddr, gp);
      }
    }
    wait_async0();
    __syncthreads();

    #pragma unroll
    for (int step = 0; step < 8; ++step) {         // K step of 4
      const int klocal = step * 4 + 2 * hf;        // lanes 16..31 hold K=2,3
      const int kglob  = kc * 32 + klocal;
      const v2f b  = *reinterpret_cast<const v2f*>(&wp_lds[mn * 260 + kglob]);
      const v2f a0 = *reinterpret_cast<const v2f*>(&x_lds[(wave * 32 + mn) * 36 + klocal]);
      const v2f a1 = *reinterpret_cast<const v2f*>(&x_lds[(wave * 32 + 16 + mn) * 36 + klocal]);
      acc0 = __builtin_amdgcn_wmma_f32_16x16x4_f32(false, a0, false, b,
                                                   (short)0, acc0, false, false);
      acc1 = __builtin_amdgcn_wmma_f32_16x16x4_f32(false, a1, false, b,
                                                   (short)0, acc1, false, false);
    }
    __syncthreads();
  }

  // scatter the 4 valid projection columns per row (D-frag: M=v+8*hf, N=mn)
  if (mn < 4) {
    const int rb = wave * 32 + 8 * hf;
    #pragma unroll
    for (int v = 0; v < 8; ++v) {
      proj_lds[(rb + v) * 4 + mn]      = acc0[v];
      proj_lds[(rb + 16 + v) * 4 + mn] = acc1[v];
    }
  }
  __syncthreads();

  // ---- per-thread: one sample through circuit + MLP ------------------------------
  const int  row   = row0 + tid;
  const bool valid = (row < Btot);     // no early return: barriers below

  float a[4];
  #pragma unroll
  for (int j = 0; j < 4; ++j)
    a[j] = tanhf(proj_lds[tid * 4 + j] + b_proj[j]) * PI_F;

  // ---- layer 1: all 16 single-qubit gates act on |0000> (product state) ----------
  // per-wire 2-vector: v = RZ(qw1).RY(qw0).RZ(a).RY(a) |0>
  float vr[4][2], vi[4][2];
  float cy1[4], sy1[4];                // sincos(a/2), reused by layer 2
  #pragma unroll
  for (int w = 0; w < 4; ++w) {
    float s1, c1;
    sincosf(0.5f * a[w], &s1, &c1);
    cy1[w] = c1; sy1[w] = s1;
    // RY(a)|0> = (c1, s1); then RZ(a): *e^{-i a/2} / *e^{+i a/2}
    const float u0r = c1 * c1, u0i = -c1 * s1;
    const float u1r = s1 * c1, u1i =  s1 * s1;
    // RY(qw[l=0,w,0])
    const float cy = qtrig[(w * 2 + 0) * 2], sy = qtrig[(w * 2 + 0) * 2 + 1];
    const float w0r = cy * u0r - sy * u1r, w0i = cy * u0i - sy * u1i;
    const float w1r = sy * u0r + cy * u1r, w1i = sy * u0i + cy * u1i;
    // RZ(qw[l=0,w,1]): bit0 *(pc,-ps), bit1 *(pc,+ps)
    const float pc = qtrig[(w * 2 + 1) * 2], ps = qtrig[(w * 2 + 1) * 2 + 1];
    vr[w][0] = w0r * pc + w0i * ps;  vi[w][0] = -w0r * ps + w0i * pc;
    vr[w][1] = w1r * pc - w1i * ps;  vi[w][1] =  w1r * ps + w1i * pc;
  }
  // tensor product: amp[b0b1b2b3] = v0[b0]*v1[b1]*v2[b2]*v3[b3] (wire0 = MSB)
  float pr01[4], pi01[4];
  #pragma unroll
  for (int b0 = 0; b0 < 2; ++b0)
    #pragma unroll
    for (int b1 = 0; b1 < 2; ++b1) {
      const int j = b0 * 2 + b1;
      pr01[j] = vr[0][b0] * vr[1][b1] - vi[0][b0] * vi[1][b1];
      pi01[j] = vr[0][b0] * vi[1][b1] + vi[0][b0] * vr[1][b1];
    }
  float pr012[8], pi012[8];
  #pragma unroll
  for (int j = 0; j < 4; ++j)
    #pragma unroll
    for (int b2 = 0; b2 < 2; ++b2) {
      const int k = j * 2 + b2;
      pr012[k] = pr01[j] * vr[2][b2] - pi01[j] * vi[2][b2];
      pi012[k] = pr01[j] * vi[2][b2] + pi01[j] * vr[2][b2];
    }
  float sr[16], si[16];
  #pragma unroll
  for (int k = 0; k < 8; ++k)
    #pragma unroll
    for (int b3 = 0; b3 < 2; ++b3) {
      const int i = k * 2 + b3;
      sr[i] = pr012[k] * vr[3][b3] - pi012[k] * vi[3][b3];
      si[i] = pr012[k] * vi[3][b3] + pi012[k] * vr[3][b3];
    }
  cnot_gate<0>(sr, si);
  cnot_gate<1>(sr, si);
  cnot_gate<2>(sr, si);

  // ---- layer 2 (state entangled: full gates) -------------------------------------
  // RZ(2a) half-angle = a -> double-angle identities from sincos(a/2)
#define ANG2(W) do {                                              \
    ry_gate<W>(sr, si, cy1[W], sy1[W]);                           \
    const float cz2 = cy1[W] * cy1[W] - sy1[W] * sy1[W];          \
    const float sz2 = 2.f * sy1[W] * cy1[W];                      \
    rz_gate<W>(sr, si, cz2, sz2); } while (0)
  ANG2(0); ANG2(1); ANG2(2); ANG2(3);
#undef ANG2
#define QW2(W) do {                                               \
    const int p0 = 8 + W * 2;                                     \
    ry_gate<W>(sr, si, qtrig[p0 * 2], qtrig[p0 * 2 + 1]);         \
    rz_gate<W>(sr, si, qtrig[(p0 + 1) * 2], qtrig[(p0 + 1) * 2 + 1]); } while (0)
  QW2(0); QW2(1); QW2(2); QW2(3);
#undef QW2
  cnot_gate<0>(sr, si);
  cnot_gate<1>(sr, si);
  cnot_gate<2>(sr, si);

  // <Z_w> expectation values
  float p[16];
  #pragma unroll
  for (int i = 0; i < 16; ++i) p[i] = sr[i] * sr[i] + si[i] * si[i];
  float z[4];
  #pragma unroll
  for (int w = 0; w < 4; ++w) {
    const int m = 8 >> w;
    float acc = 0.f;
    #pragma unroll
    for (int i = 0; i < 16; ++i) acc += (i & m) ? -p[i] : p[i];
    z[w] = acc;
  }

  // tiny MLP head (uniform weight addresses -> scalar-cached loads)
  float h[16];
  #pragma unroll
  for (int i = 0; i < 16; ++i) {
    float acc = b_h[i];
    #pragma unroll
    for (int j = 0; j < 4; ++j) acc = fmaf(z[j], w_h[i * 4 + j], acc);
    h[i] = fmaxf(acc, 0.f);
  }
  float o[10];
  #pragma unroll
  for (int k = 0; k < 10; ++k) {
    float acc = b_out[k];
    #pragma unroll
    for (int i = 0; i < 16; ++i) acc = fmaf(h[i], w_out[k * 16 + i], acc);
    o[k] = acc;
  }

  // ---- stage results in LDS (x_lds is free now) and write coalesced --------------
  #pragma unroll
  for (int k = 0; k < 10; ++k)
    x_lds[tid * 10 + k] = valid ? o[k] : 0.f;
  __syncthreads();

  const int nvalid = (Btot - row0 < 256) ? (Btot - row0) : 256;
  const int limit  = nvalid * 10;
  for (int idx = tid; idx < 2560; idx += 256) {
    if (idx < limit) out[(size_t)row0 * 10 + idx] = x_lds[idx];
  }
}

extern "C" void kernel_launch(void* const* d_in, const int* in_sizes, int n_in,
                              void* d_out, int out_size, void* d_ws, size_t ws_size,
                              hipStream_t stream) {
  const float* x      = (const float*)d_in[0];
  const float* w_proj = (const float*)d_in[1];
  const float* b_proj = (const float*)d_in[2];
  const float* qw     = (const float*)d_in[3];
  const float* w_h    = (const float*)d_in[4];
  const float* b_h    = (const float*)d_in[5];
  const float* w_out  = (const float*)d_in[6];
  const float* b_out  = (const float*)d_in[7];
  float* out = (float*)d_out;

  const int Btot = in_sizes[0] / 256;          // x is [B,256]
  const int blocks = (Btot + 255) / 256;       // 256 rows per block
  QuantumHead_80925773791384_kernel<<<blocks, 256, 0, stream>>>(
      x, w_proj, b_proj, qw, w_h, b_h, w_out, b_out, out, Btot);
}